// ContourChamferLoss_35098472743452
// MI455X (gfx1250) — compile-verified
//
#include <hip/hip_runtime.h>

// CDNA5 (gfx1250) Chamfer loss:
//   d2(i,j) = |pc_i|^2 + |ref_j|^2 - 2 pc_i . ref_j
// cross term via V_WMMA_F32_16X16X4_F32 (K=2 zero-padded to 4),
// min over sqrt == sqrt over min (monotone), so all reductions on d2,
// clamped >=0 so float mins can use integer atomics on bit-punned floats.
// Inner loop is software-pipelined: ref point for tile j+1 is loaded while
// tile j runs on the matrix pipe, so s_wait_loadcnt hides under a full tile.

typedef float v2f __attribute__((ext_vector_type(2)));
typedef float v8f __attribute__((ext_vector_type(8)));

#define INF_BITS 0x7F800000u

__device__ __forceinline__ float    u2f(unsigned u) { return __uint_as_float(u); }
__device__ __forceinline__ unsigned f2u(float f)    { return __float_as_uint(f); }

__global__ void chamfer_init(unsigned* __restrict__ colMin, float* __restrict__ out, int M) {
    int idx = blockIdx.x * blockDim.x + threadIdx.x;
    if (idx < M) colMin[idx] = INF_BITS;
    if (idx == 0) out[0] = 0.0f;
}

__global__ __launch_bounds__(256)
void chamfer_main(const float2* __restrict__ pc, const float2* __restrict__ ref,
                  float* __restrict__ out, unsigned* __restrict__ colMinG, int M) {
    __shared__ unsigned colMinLds[16384];   // 64 KB, M == 16384 per reference

    const int tid  = threadIdx.x;
    const int lane = tid & 31;
    const int wave = tid >> 5;
    const int half = lane >> 4;   // 0: rows M=0..7 & K=0,1 ; 1: rows M=8..15 & K=2,3 (pad)
    const int lr   = lane & 15;

    for (int i = tid; i < M; i += 256) colMinLds[i] = INF_BITS;
    __syncthreads();

    // ---- A tile: this wave owns rows [rowBase, rowBase+16) ----
    const int rowBase = blockIdx.x * 128 + wave * 16;
    float2 ap = pc[rowBase + lr];                       // lanes 0..15 carry the 16 rows
    float  rn = ap.x * ap.x + ap.y * ap.y;              // |pc_row|^2 at lane == row-lr

    float an[8];                                        // per-half row norms (rows 8*half+v)
#pragma unroll
    for (int v = 0; v < 8; ++v) an[v] = __shfl(rn, 8 * half + v, 32);

    // Fold the -2 of the cross term into A.  Upper half-wave = K=2,3 zero-pad.
    v2f A;
    A.x = (half == 0) ? (-2.0f * ap.x) : 0.0f;
    A.y = (half == 0) ? (-2.0f * ap.y) : 0.0f;
    v8f Cz = {};                                        // zero accumulator

    float rowAcc[8];
#pragma unroll
    for (int v = 0; v < 8; ++v) rowAcc[v] = u2f(INF_BITS);

    const int ntiles = M >> 4;

    // Software pipeline: rp holds tile j's ref point, rpn preloads tile j+1.
    float2 rp = ref[lr];
#pragma unroll 2
    for (int j = 0; j < ntiles; ++j) {
        const int jn = (j + 1 < ntiles) ? (j + 1) : 0;  // clamp to stay in-bounds
        float2 rpn = ref[lr + 16 * jn];                 // issue early; consumed next iter

        float bn = rp.x * rp.x + rp.y * rp.y;           // |ref_col|^2

        v2f B;                                          // B mirrors A layout; K=2,3 pad = 0
        B.x = (half == 0) ? rp.x : 0.0f;
        B.y = (half == 0) ? rp.y : 0.0f;

        // D[v] = -2 * (pc_row . ref_col) for row = 8*half+v, col = lr (tile-local)
        v8f D = __builtin_amdgcn_wmma_f32_16x16x4_f32(
            /*neg_a=*/false, A, /*neg_b=*/false, B,
            /*c_mod=*/(short)0, Cz, /*reuse_a=*/false, /*reuse_b=*/false);

        float cmin = u2f(INF_BITS);
#pragma unroll
        for (int v = 0; v < 8; ++v) {
            float d2 = fmaxf(an[v] + bn + D[v], 0.0f);  // clamp: fp error + enables int mins
            rowAcc[v] = fminf(rowAcc[v], d2);
            cmin      = fminf(cmin, d2);
        }
        // all 32 lanes issue the LDS atomic (2 lanes/address): no bpermute round-trip,
        // no exec-mask branch in the hot loop.
        atomicMin(&colMinLds[lr + 16 * j], f2u(cmin));  // ds_min_u32, valid: all >= 0

        rp = rpn;
    }

    // ---- row mins: reduce across the 16 lanes of each half ----
#pragma unroll
    for (int off = 1; off < 16; off <<= 1) {
#pragma unroll
        for (int v = 0; v < 8; ++v)
            rowAcc[v] = fminf(rowAcc[v], __shfl_xor(rowAcc[v], off, 32));
    }
    if (lr == 0) {                                      // lanes 0 and 16: rows 0-7 / 8-15
        float s = 0.0f;
#pragma unroll
        for (int v = 0; v < 8; ++v) s += sqrtf(rowAcc[v]);
        atomicAdd(out, s);
    }

    __syncthreads();
    // ---- merge this block's column mins into global workspace ----
    for (int i = tid; i < M; i += 256)
        atomicMin(&colMinG[i], colMinLds[i]);           // global_atomic_min_u32
}

__global__ __launch_bounds__(256)
void chamfer_colsum(const unsigned* __restrict__ colMinG, float* __restrict__ out, int M) {
    __shared__ float waveSums[8];
    float s = 0.0f;
    for (int i = threadIdx.x; i < M; i += 256)
        s += sqrtf(u2f(colMinG[i]));
#pragma unroll
    for (int off = 16; off >= 1; off >>= 1) s += __shfl_xor(s, off, 32);
    if ((threadIdx.x & 31) == 0) waveSums[threadIdx.x >> 5] = s;
    __syncthreads();
    if (threadIdx.x == 0) {
        float t = 0.0f;
        for (int w = 0; w < 8; ++w) t += waveSums[w];
        atomicAdd(out, t);
    }
}

extern "C" void kernel_launch(void* const* d_in, const int* in_sizes, int n_in,
                              void* d_out, int out_size, void* d_ws, size_t ws_size,
                              hipStream_t stream) {
    const float2* pc  = (const float2*)d_in[0];   // 128*128 points, flattened (16384, 2)
    const float2* ref = (const float2*)d_in[1];   // (16384, 2)
    float* out = (float*)d_out;

    const int Np = in_sizes[0] / 2;               // 16384 render points
    const int M  = in_sizes[1] / 2;               // 16384 ref points

    unsigned* colMin = (unsigned*)d_ws;           // M u32 (64 KB) of squared-dist bits

    chamfer_init<<<(M + 255) / 256, 256, 0, stream>>>(colMin, out, M);
    chamfer_main<<<Np / 128, 256, 0, stream>>>(pc, ref, out, colMin, M);
    chamfer_colsum<<<1, 256, 0, stream>>>(colMin, out, M);
}